// RegionLoss_78391743086895
// MI455X (gfx1250) — compile-verified
//
#include <hip/hip_runtime.h>
#include <math.h>

typedef __attribute__((ext_vector_type(2))) float v2f;
typedef __attribute__((ext_vector_type(8))) float v8f;

#define KK     9
#define MAX_GT 5
#define LL     21          // 2*K+3
#define BB     128
#define NH     76
#define NW     76
#define CH     20          // 2*K+1+NC
#define PLANE  (NH*NW)     // 5776
#define NCONF  (BB*PLANE)  // 739328

#define NB1 304
#define NT1 256
#define NB2 3
#define NT2 256

__device__ __forceinline__ float sigmoidf_(float v) {
    return 1.0f / (1.0f + __expf(-v));
}

// ---------------------------------------------------------------------------
// K0: per-(b,g) GT cell table: cell index (cy*NW+cx) if valid & in-range, else -1
// ---------------------------------------------------------------------------
__global__ void gt_cells_kernel(const float* __restrict__ target,
                                int* __restrict__ cells) {
    int t = blockIdx.x * blockDim.x + threadIdx.x;
    if (t >= BB * MAX_GT) return;
    int b = t / MAX_GT, g = t % MAX_GT;
    const float* tb = target + b * (MAX_GT * LL);
    bool valid = true;
    for (int j = 0; j <= g; ++j) valid = valid && (tb[j * LL + 1] != 0.0f);
    int cell = -1;
    if (valid) {
        float gx = tb[g * LL + 1] * (float)NW;
        float gy = tb[g * LL + 2] * (float)NH;
        int cx = (int)gx, cy = (int)gy;
        if (cx >= 0 && cx < NW && cy >= 0 && cy < NH) cell = cy * NW + cx;
    }
    cells[t] = cell;
}

// ---------------------------------------------------------------------------
// K1: loss_conf + nProposals over the conf channel (channel 2K=18).
// Squared errors are folded 64-at-a-time per wave through
// V_WMMA_F32_16X16X4_F32 with an all-ones A matrix (pure f32 reduction).
// ---------------------------------------------------------------------------
__global__ __launch_bounds__(NT1)
void conf_kernel(const float* __restrict__ output,
                 const int* __restrict__ cells,
                 float* __restrict__ partials) {
    const int tid   = threadIdx.x;
    const int lane  = tid & 31;
    const int wave  = tid >> 5;
    const int gwave = blockIdx.x * (NT1 / 32) + wave;
    const int nwav  = NB1 * (NT1 / 32);
    const int nChunks = (NCONF + 63) >> 6;

    v8f acc = {0.f, 0.f, 0.f, 0.f, 0.f, 0.f, 0.f, 0.f};
    v2f ones = {1.0f, 1.0f};
    float prop = 0.0f;

    for (int chunk = gwave; chunk < nChunks; chunk += nwav) {
        int base = chunk << 6;
        float e2v[2];
#pragma unroll
        for (int u = 0; u < 2; ++u) {
            int idx  = base + u * 32 + lane;
            bool in  = idx < NCONF;
            int cidx = in ? idx : (NCONF - 1);
            int b  = cidx / PLANE;
            int hw = cidx - b * PLANE;
            float v = output[(size_t)b * (CH * PLANE) + (2 * KK) * PLANE + hw];
            float s = sigmoidf_(v);
            float t = 0.0f;
#pragma unroll
            for (int g = 0; g < MAX_GT; ++g)
                t = (cells[b * MAX_GT + g] == hw) ? 1.0f : t;
            float e = s - t;
            e2v[u] = in ? e * e : 0.0f;
            prop  += (in && s > 0.5f) ? 1.0f : 0.0f;
        }
        v2f bv = {e2v[0], e2v[1]};
        // D = ones(16x4) * B(4x16) + C : every D column n accumulates the sum
        // of its 4 B slots; 64 unique f32 values consumed per instruction.
        acc = __builtin_amdgcn_wmma_f32_16x16x4_f32(
            false, ones, false, bv, (short)0, acc, false, false);
    }

    // all D rows identical -> wave total = sum over lanes 0..15 of acc[0]
    float csum = (lane < 16) ? acc[0] : 0.0f;

    __shared__ float red0[NT1];
    __shared__ float red1[NT1];
    red0[tid] = csum;
    red1[tid] = prop;
    __syncthreads();
    for (int s = NT1 / 2; s > 0; s >>= 1) {
        if (tid < s) { red0[tid] += red0[tid + s]; red1[tid] += red1[tid + s]; }
        __syncthreads();
    }
    if (tid == 0) {
        partials[blockIdx.x * 2 + 0] = red0[0];
        partials[blockIdx.x * 2 + 1] = red1[0];
    }
}

// ---------------------------------------------------------------------------
// K2: per-GT scattered work: loss_x, loss_y, nGT, nCorrect (640 live threads)
// ---------------------------------------------------------------------------
__global__ __launch_bounds__(NT2)
void gt_kernel(const float* __restrict__ output,
               const float* __restrict__ target,
               float* __restrict__ partials) {
    int tid = threadIdx.x;
    int t = blockIdx.x * NT2 + tid;
    float lx = 0.f, ly = 0.f, ngt = 0.f, ncorr = 0.f;
    if (t < BB * MAX_GT) {
        int b = t / MAX_GT, g = t % MAX_GT;
        const float* tb = target + b * (MAX_GT * LL);
        bool valid = true;
        for (int j = 0; j <= g; ++j) valid = valid && (tb[j * LL + 1] != 0.0f);

        float gx0 = tb[g * LL + 1] * (float)NW;
        float gy0 = tb[g * LL + 2] * (float)NH;
        int cx = (int)gx0, cy = (int)gy0;
        int cxg = cx < 0 ? 0 : (cx > NW - 1 ? NW - 1 : cx);
        int cyg = cy < 0 ? 0 : (cy > NH - 1 ? NH - 1 : cy);
        bool inr = (cx >= 0 && cx < NW && cy >= 0 && cy < NH);
        const float* base = output + (size_t)b * (CH * PLANE) + cyg * NW + cxg;
        float cxf = (float)cx, cyf = (float)cy;
        float denom = __expf(2.0f) - 1.0f + 1e-5f;
        float csum = 0.0f;
#pragma unroll
        for (int k = 0; k < KK; ++k) {
            float gxk = tb[g * LL + 1 + 2 * k];
            float gyk = tb[g * LL + 2 + 2 * k];
            float rx = base[(2 * k) * PLANE];
            float ry = base[(2 * k + 1) * PLANE];
            float xk = (k == 0) ? sigmoidf_(rx) : rx;
            float yk = (k == 0) ? sigmoidf_(ry) : ry;
            // loss at the (unclipped) GT cell; valid => clipped == unclipped
            float tx = gxk * (float)NW - cxf;
            float ty = gyk * (float)NH - cyf;
            if (valid && inr) {
                float dx = xk - tx, dy = yk - ty;
                lx += dx * dx;
                ly += dy * dy;
            }
            // corner confidence (uses clipped cell, matches reference)
            float prx = (xk + cxf) * (1.0f / (float)NW);
            float pry = (yk + cyf) * (1.0f / (float)NH);
            float ddx = (gxk - prx) * 640.0f;
            float ddy = (gyk - pry) * 480.0f;
            float d = sqrtf(ddx * ddx + ddy * ddy);
            float c = (d < 80.0f)
                        ? (__expf(2.0f * (1.0f - d * (1.0f / 80.0f))) - 1.0f) / denom
                        : 0.0f;
            csum += c;
        }
        float conf_gt = csum * (1.0f / (float)KK);
        ngt = valid ? 1.0f : 0.0f;
        ncorr = (valid && conf_gt > 0.7f) ? 1.0f : 0.0f;
    }

    __shared__ float red[NT2];
    float vals[4] = {lx, ly, ngt, ncorr};
    for (int q = 0; q < 4; ++q) {
        red[tid] = vals[q];
        __syncthreads();
        for (int s = NT2 / 2; s > 0; s >>= 1) {
            if (tid < s) red[tid] += red[tid + s];
            __syncthreads();
        }
        if (tid == 0) partials[blockIdx.x * 4 + q] = red[0];
        __syncthreads();
    }
}

// ---------------------------------------------------------------------------
// K3: deterministic fixed-order final combine -> 7 outputs
// ---------------------------------------------------------------------------
__global__ void finalize_kernel(const float* __restrict__ confp,
                                const float* __restrict__ gtp,
                                float* __restrict__ out) {
    if (threadIdx.x != 0 || blockIdx.x != 0) return;
    float lc = 0.f, np = 0.f;
    for (int i = 0; i < NB1; ++i) { lc += confp[2 * i]; np += confp[2 * i + 1]; }
    float lx = 0.f, ly = 0.f, ngt = 0.f, ncorr = 0.f;
    for (int i = 0; i < NB2; ++i) {
        lx    += gtp[4 * i + 0];
        ly    += gtp[4 * i + 1];
        ngt   += gtp[4 * i + 2];
        ncorr += gtp[4 * i + 3];
    }
    out[0] = lx + ly + lc;            // loss
    out[1] = ngt * (1.0f / BB);       // nGT
    out[2] = ncorr * (1.0f / BB);     // nCorrect
    out[3] = np * (1.0f / BB);        // nProposals
    out[4] = lx;                      // loss_x
    out[5] = ly;                      // loss_y
    out[6] = lc;                      // loss_conf
}

extern "C" void kernel_launch(void* const* d_in, const int* in_sizes, int n_in,
                              void* d_out, int out_size, void* d_ws, size_t ws_size,
                              hipStream_t stream) {
    const float* output = (const float*)d_in[0];
    const float* target = (const float*)d_in[1];
    float* out = (float*)d_out;

    int*   cells = (int*)d_ws;                                   // 640 ints
    float* confp = (float*)((char*)d_ws + BB * MAX_GT * sizeof(int)); // NB1*2 floats
    float* gtp   = confp + NB1 * 2;                              // NB2*4 floats

    gt_cells_kernel<<<1, BB * MAX_GT, 0, stream>>>(target, cells);
    conf_kernel<<<NB1, NT1, 0, stream>>>(output, cells, confp);
    gt_kernel<<<NB2, NT2, 0, stream>>>(output, target, gtp);
    finalize_kernel<<<1, 32, 0, stream>>>(confp, gtp, out);
}